// TrainedLiftPDEGCNN_87557203296304
// MI455X (gfx1250) — compile-verified
//
#include <hip/hip_runtime.h>
#include <math.h>

#define OO   8
#define CC   16
#define KS   7
#define KK   49
#define MK   125
#define NB   4
#define HH   192
#define WW   192
#define HWH  (HH*WW)
#define OHW  (OO*HH*WW)
#define TEN  ((size_t)NB*CC*OHW)
#define NLYR 4
#define BIGV 1e30f
#define EPSV 1e-5f
#define PI_F 3.14159265358979f

typedef __attribute__((ext_vector_type(2))) float v2f;
typedef __attribute__((ext_vector_type(8))) float v8f;

// ---------------------------------------------------------------- setup
__global__ void k_setup(const float* __restrict__ lw, float* R, float* wr,
                        int* mDo, int* mDy, int* mDx, float* mFy, float* mFx,
                        float* mC1, float* mC2, float* mC3) {
  int tid = threadIdx.x;
  // morph geometry (independent of R)
  for (int j = tid; j < MK; j += blockDim.x) {
    int dt = j/25 - 2, dy = (j/5)%5 - 2, dx = j%5 - 2;
    float dth = 2.f*PI_F/OO;
    float a = dt*dth;
    mDo[j] = dt;
    mC1[j] =  cosf(a*0.5f)*dx + sinf(a*0.5f)*dy;
    mC2[j] = -sinf(a*0.5f)*dx + cosf(a*0.5f)*dy;
    mC3[j] = a;
    for (int o = 0; o < OO; ++o) {
      float th = o*dth;
      float gx = cosf(th)*dx - sinf(th)*dy;
      float gy = sinf(th)*dx + cosf(th)*dy;
      float fx0 = floorf(gx), fy0 = floorf(gy);
      mDx[j*OO+o] = (int)fx0; mFx[j*OO+o] = gx - fx0;
      mDy[j*OO+o] = (int)fy0; mFy[j*OO+o] = gy - fy0;
    }
  }
  // rotation matrices R[o][q][p] (each (o,q) row owned by one thread)
  for (int t = tid; t < OO*KK; t += blockDim.x) {
    int o = t/KK, q = t%KK;
    float* row = R + (size_t)t*KK;
    for (int p = 0; p < KK; ++p) row[p] = 0.f;
    float cen = (KS-1)*0.5f;
    float ys = (float)(q/KS) - cen, xs = (float)(q%KS) - cen;
    float th = 2.f*PI_F*o/OO;
    float sy =  cosf(th)*ys + sinf(th)*xs + cen;
    float sx = -sinf(th)*ys + cosf(th)*xs + cen;
    float y0 = floorf(sy), x0 = floorf(sx);
    float fy = sy-y0, fx = sx-x0;
    for (int dy=0; dy<2; ++dy) for (int dx=0; dx<2; ++dx) {
      int yi = (int)y0+dy, xi = (int)x0+dx;
      if (yi>=0 && yi<KS && xi>=0 && xi<KS)
        row[yi*KS+xi] += (dy?fy:1.f-fy)*(dx?fx:1.f-fx);
    }
  }
  __syncthreads();
  // wr[c,o,ci,q] = sum_p R[o,q,p] * lw[c,ci,p]
  for (int t = tid; t < CC*OO*3*KK; t += blockDim.x) {
    int q = t % KK; int r = t / KK;
    int ci = r % 3; r /= 3;
    int o = r % OO; int c = r / OO;
    const float* Rr = R  + ((size_t)o*KK + q)*KK;
    const float* wv = lw + ((size_t)c*3 + ci)*KK;
    float s = 0.f;
    for (int p=0;p<KK;++p) s += Rr[p]*wv[p];
    wr[t] = s;
  }
}

// ---------------------------------------------------------------- lift conv
__global__ void __launch_bounds__(256) k_lift(const float* __restrict__ x,
                                              const float* __restrict__ wr,
                                              float* __restrict__ out) {
  __shared__ float sW[3*KK];
  __shared__ float sX[3][22][22];
  int tid = threadIdx.x;
  int bz = blockIdx.z;                    // bz = (b*CC + c)*OO + o
  int b  = bz / (CC*OO);
  int co = bz % (CC*OO);
  int h0 = blockIdx.y*16, w0 = blockIdx.x*16;
  if (tid < 3*KK) sW[tid] = wr[(size_t)co*3*KK + tid];
  for (int t = tid; t < 3*484; t += blockDim.x) {
    int ci = t/484; int rem = t%484; int ly = rem/22; int lx = rem%22;
    int gi = h0 + ly - 3; int gj = w0 + lx - 3;      // reflect pad 3
    gi = gi < 0 ? -gi : (gi >= HH ? 2*HH-2-gi : gi);
    gj = gj < 0 ? -gj : (gj >= WW ? 2*WW-2-gj : gj);
    sX[ci][ly][lx] = x[((size_t)(b*3+ci)*HH + gi)*WW + gj];
  }
  __syncthreads();
  int ty = tid>>4, tx = tid&15;
  float acc = 0.f;
  for (int ci=0; ci<3; ++ci)
    for (int ky=0; ky<KS; ++ky)
      for (int kx=0; kx<KS; ++kx)
        acc += sW[ci*KK + ky*KS + kx] * sX[ci][ty+ky][tx+kx];
  out[(size_t)bz*HWH + (size_t)(h0+ty)*WW + (w0+tx)] = acc;
}

// ---------------------------------------------------------------- batchnorm
__global__ void k_zero_stats(float* stats) {
  if (threadIdx.x < 2*CC) stats[threadIdx.x] = 0.f;
}

__global__ void __launch_bounds__(256) k_bn_reduce(const float* __restrict__ v,
                                                   float* __restrict__ stats) {
  __shared__ float sS[256], sQ[256];
  int tid = threadIdx.x;
  int c = blockIdx.y, b = blockIdx.z;
  size_t base = ((size_t)b*CC + c)*OHW + (size_t)blockIdx.x*16384;
  float s=0.f, q=0.f;
  for (int i=0;i<64;++i){ float xv = v[base + (size_t)i*256 + tid]; s += xv; q += xv*xv; }
  sS[tid]=s; sQ[tid]=q; __syncthreads();
  for (int st=128; st>0; st>>=1){
    if (tid<st){ sS[tid]+=sS[tid+st]; sQ[tid]+=sQ[tid+st]; }
    __syncthreads();
  }
  if (tid==0){ atomicAdd(&stats[c], sS[0]); atomicAdd(&stats[CC+c], sQ[0]); }
}

__global__ void __launch_bounds__(256) k_bn_apply(const float* __restrict__ src,
                                                  float* __restrict__ dst,
                                                  const float* __restrict__ stats,
                                                  const float* __restrict__ gamma,
                                                  const float* __restrict__ beta) {
  size_t idx = (size_t)blockIdx.x*blockDim.x + threadIdx.x;
  if (idx >= TEN) return;
  int c = (int)((idx / OHW) % CC);
  float n = (float)((size_t)NB*OHW);
  float m = stats[c]/n;
  float var = stats[CC+c]/n - m*m;
  dst[idx] = (src[idx]-m)*rsqrtf(var+EPSV)*gamma[c] + beta[c];
}

// ---------------------------------------------------------------- convect
__global__ void __launch_bounds__(256) k_convect(const float* __restrict__ u,
                                                 const float* __restrict__ g,
                                                 float* __restrict__ out) {
  size_t idx = (size_t)blockIdx.x*blockDim.x + threadIdx.x;
  if (idx >= TEN) return;
  int wq = (int)(idx % WW); size_t t = idx / WW;
  int hq = (int)(t % HH); t /= HH;
  int o  = (int)(t % OO); t /= OO;
  int c  = (int)(t % CC); int b = (int)(t / CC);
  float gx = g[c*3+0], gy = g[c*3+1], gth = g[c*3+2];
  float th = 2.f*PI_F*o/OO;
  float sx = cosf(th)*gx - sinf(th)*gy;
  float sy = sinf(th)*gx + cosf(th)*gy;
  float so = gth / (2.f*PI_F/OO);
  float osrc = (float)o - so;
  float o0f = floorf(osrc); float fo = osrc - o0f;
  float y0f = floorf(-sy);  float fy = -sy - y0f;
  float x0f = floorf(-sx);  float fx = -sx - x0f;
  int o0 = (int)o0f, y0 = (int)y0f, x0 = (int)x0f;
  const float* base = u + ((size_t)b*CC + c)*OHW;
  float acc = 0.f;
  for (int dd=0; dd<2; ++dd) {
    int oi = ((o0+dd)%OO + OO)%OO;
    float wo = dd ? fo : 1.f-fo;
    for (int dy=0; dy<2; ++dy) {
      int yi = hq + y0 + dy; yi = yi<0?0:(yi>=HH?HH-1:yi);
      float wy = wo*(dy?fy:1.f-fy);
      for (int dx=0; dx<2; ++dx) {
        int xi = wq + x0 + dx; xi = xi<0?0:(xi>=WW?WW-1:xi);
        acc += wy*(dx?fx:1.f-fx) * base[(size_t)oi*HWH + (size_t)yi*WW + xi];
      }
    }
  }
  out[idx] = acc;
}

// ------------------------------------------- morph (async-to-LDS tiled)
__global__ void __launch_bounds__(256) k_morph(const float* __restrict__ v,
                                               const float* __restrict__ metric,
                                               const int* __restrict__ mDo,
                                               const int* __restrict__ mDy,
                                               const int* __restrict__ mDx,
                                               const float* __restrict__ mFy,
                                               const float* __restrict__ mFx,
                                               const float* __restrict__ mC1,
                                               const float* __restrict__ mC2,
                                               const float* __restrict__ mC3,
                                               float* __restrict__ out, int dilate) {
  __shared__ float sP[5][22][22];
  __shared__ float sKern[MK], sFyL[MK], sFxL[MK];
  __shared__ int   sDyL[MK], sDxL[MK], sPlL[MK];
  int tid = threadIdx.x;
  int bz = blockIdx.z;                    // (b*CC + c)*OO + o
  int o  = bz % OO;
  int bc = bz / OO;
  int c  = bc % CC;
  int h0 = blockIdx.y*16, w0 = blockIdx.x*16;
  float PV = dilate ? -BIGV : BIGV;

  // Boundary blocks: prefill tile with pad value, commit before async writes.
  bool interior = (h0 >= 3) && (h0 + 19 <= HH) && (w0 >= 3) && (w0 + 19 <= WW);
  if (!interior) {
    for (int t = tid; t < 5*484; t += blockDim.x) {
      int pl = t/484; int rem = t%484;
      sP[pl][rem/22][rem%22] = PV;
    }
    asm volatile("s_wait_dscnt 0x0" ::: "memory");
  }
  // Direct memory -> LDS staging via CDNA5 async loads (ASYNCcnt-tracked).
  for (int t = tid; t < 5*484; t += blockDim.x) {
    int pl = t/484; int rem = t%484; int ly = rem/22, lx = rem%22;
    int oi = (o + pl - 2 + OO) % OO;
    int gy = h0 + ly - 3, gx = w0 + lx - 3;
    if (gy>=0 && gy<HH && gx>=0 && gx<WW) {
      const float* gp = v + ((size_t)bc*OO + oi)*HWH + (size_t)gy*WW + gx;
      unsigned lds_off = (unsigned)(uintptr_t)(&sP[pl][ly][lx]);
      asm volatile("global_load_async_to_lds_b32 %0, %1, off"
                   :: "v"(lds_off), "v"((unsigned long long)(uintptr_t)gp)
                   : "memory");
    }
  }
  if (tid < MK) {
    int j = tid;
    sPlL[j] = mDo[j] + 2;
    sDyL[j] = mDy[j*OO+o]; sFyL[j] = mFy[j*OO+o];
    sDxL[j] = mDx[j*OO+o]; sFxL[j] = mFx[j*OO+o];
    float m0 = metric[c*3+0], m1 = metric[c*3+1], m2 = metric[c*3+2];
    float c1 = mC1[j], c2 = mC2[j], c3 = mC3[j];
    float rho2 = m0*c1*c1 + m1*c2*c2 + m2*c3*c3;
    float expo = 13.f/3.f;
    float nu = 0.3f * powf(1.3f, -expo);
    sKern[j] = nu * powf(rho2, expo*0.5f);
  }
  asm volatile("s_wait_asynccnt 0x0" ::: "memory");
  __syncthreads();

  int ty = tid>>4, tx = tid&15;
  float acc = PV;
  for (int j=0;j<MK;++j) {
    int pl = sPlL[j];
    int yy = ty + 3 + sDyL[j], xx = tx + 3 + sDxL[j];
    float fy = sFyL[j], fx = sFxL[j];
    float v00 = sP[pl][yy][xx],   v01 = sP[pl][yy][xx+1];
    float v10 = sP[pl][yy+1][xx], v11 = sP[pl][yy+1][xx+1];
    float val = (1.f-fy)*((1.f-fx)*v00 + fx*v01) + fy*((1.f-fx)*v10 + fx*v11);
    float kj = sKern[j];
    acc = dilate ? fmaxf(acc, val - kj) : fminf(acc, val + kj);
  }
  out[((size_t)bc*OO + o)*HWH + (size_t)(h0+ty)*WW + (w0+tx)] = acc;
}

// ------------------------------------------------- channel mix via fp32 WMMA
// z[d, pix] = sum_{k=0..31} lin[d,k] * cat(dil,ero)[k, pix]
// One wave handles one 16-pixel strip; K=32 via 8x V_WMMA_F32_16X16X4_F32.
__global__ void __launch_bounds__(256) k_mix_wmma(const float* __restrict__ dil,
                                                  const float* __restrict__ ero,
                                                  const float* __restrict__ lin,
                                                  float* __restrict__ z) {
  int tid = threadIdx.x;
  int wave = tid >> 5, lane = tid & 31;
  int strip = blockIdx.x*8 + wave;
  size_t q0 = (size_t)strip * 16;        // flat (b,o,h,w) pixel base
  int b = (int)(q0 / OHW);
  size_t r0 = q0 % OHW;
  int pix = lane & 15, half = lane >> 4;
  size_t r = r0 + (size_t)pix;
  v8f acc = {0.f,0.f,0.f,0.f,0.f,0.f,0.f,0.f};
  for (int k0 = 0; k0 < 32; k0 += 4) {
    int kA = k0 + half*2;                // A: lanes 0-15 K={k0,k0+1}, lanes 16-31 K={k0+2,k0+3}
    v2f av, bv;
    av.x = lin[pix*32 + kA];
    av.y = lin[pix*32 + kA + 1];
    const float* s0 = (kA   < CC) ? (dil + ((size_t)b*CC + kA)*OHW)
                                  : (ero + ((size_t)b*CC + (kA-CC))*OHW);
    const float* s1 = (kA+1 < CC) ? (dil + ((size_t)b*CC + kA+1)*OHW)
                                  : (ero + ((size_t)b*CC + (kA+1-CC))*OHW);
    bv.x = s0[r];
    bv.y = s1[r];
    acc = __builtin_amdgcn_wmma_f32_16x16x4_f32(
        /*neg_a=*/false, av, /*neg_b=*/false, bv,
        /*c_mod=*/(short)0, acc, /*reuse_a=*/false, /*reuse_b=*/false);
  }
  size_t zb = (size_t)b*CC*OHW;
#pragma unroll
  for (int r8=0; r8<8; ++r8) {
    int d = half*8 + r8;                 // D layout: VGPR r -> row r (lanes<16) / r+8
    z[zb + (size_t)d*OHW + r] = acc[r8];
  }
}

// ---------------------------------------------------------------- head
__global__ void __launch_bounds__(256) k_head(const float* __restrict__ u,
                                              const float* __restrict__ fw,
                                              float* __restrict__ out) {
  int idx = blockIdx.x*blockDim.x + threadIdx.x;
  if (idx >= NB*HWH) return;
  int b = idx / HWH; int rr = idx % HWH;
  float acc = 0.f;
  for (int c=0;c<CC;++c) {
    const float* base = u + ((size_t)b*CC + c)*OHW + rr;
    float m = -BIGV;
    for (int o=0;o<OO;++o) m = fmaxf(m, base[(size_t)o*HWH]);
    acc += fw[c]*m;
  }
  out[idx] = 1.f/(1.f+expf(-acc));
}

// ---------------------------------------------------------------- launch
extern "C" void kernel_launch(void* const* d_in, const int* in_sizes, int n_in,
                              void* d_out, int out_size, void* d_ws, size_t ws_size,
                              hipStream_t stream) {
  (void)in_sizes; (void)n_in; (void)out_size; (void)ws_size;
  const float* x       = (const float*)d_in[0];
  const float* lift_w  = (const float*)d_in[1];
  const float* lift_g  = (const float*)d_in[2];
  const float* lift_b  = (const float*)d_in[3];
  const float* pde_g   = (const float*)d_in[4];
  const float* pde_met = (const float*)d_in[5];
  const float* pde_lin = (const float*)d_in[6];
  const float* pde_ga  = (const float*)d_in[7];
  const float* pde_be  = (const float*)d_in[8];
  const float* final_w = (const float*)d_in[9];
  float* out = (float*)d_out;

  float* W    = (float*)d_ws;
  float* bufU = W;
  float* bufA = W + TEN;
  float* bufB = W + 2*TEN;
  float* bufC = W + 3*TEN;
  float* tab  = W + 4*TEN;
  float* wr  = tab;  tab += CC*OO*3*KK;    // 18816
  float* Rt  = tab;  tab += OO*KK*KK;      // 19208
  float* mC1 = tab;  tab += MK;
  float* mC2 = tab;  tab += MK;
  float* mC3 = tab;  tab += MK;
  float* mFy = tab;  tab += MK*OO;
  float* mFx = tab;  tab += MK*OO;
  int* mDo = (int*)tab; tab += MK;
  int* mDy = (int*)tab; tab += MK*OO;
  int* mDx = (int*)tab; tab += MK*OO;
  float* stats = tab;   tab += 2*CC;

  dim3 tile(12,12,NB*CC*OO);
  int nElemBlocks = (int)(TEN/256);

  k_setup<<<1,256,0,stream>>>(lift_w, Rt, wr, mDo, mDy, mDx, mFy, mFx, mC1, mC2, mC3);
  k_lift<<<tile,256,0,stream>>>(x, wr, bufU);
  k_zero_stats<<<1,64,0,stream>>>(stats);
  k_bn_reduce<<<dim3(18,CC,NB),256,0,stream>>>(bufU, stats);
  k_bn_apply<<<nElemBlocks,256,0,stream>>>(bufU, bufU, stats, lift_g, lift_b);

  for (int i=0;i<NLYR;++i) {
    const float* g0 = pde_g   + ((size_t)i*2+0)*CC*3;
    const float* g1 = pde_g   + ((size_t)i*2+1)*CC*3;
    const float* m0 = pde_met + ((size_t)i*2+0)*CC*3;
    const float* m1 = pde_met + ((size_t)i*2+1)*CC*3;
    const float* li = pde_lin + (size_t)i*CC*2*CC;
    const float* ga = pde_ga  + (size_t)i*CC;
    const float* be = pde_be  + (size_t)i*CC;
    k_convect<<<nElemBlocks,256,0,stream>>>(bufU, g0, bufA);
    k_morph<<<tile,256,0,stream>>>(bufA, m0, mDo,mDy,mDx,mFy,mFx,mC1,mC2,mC3, bufB, 1);
    k_convect<<<nElemBlocks,256,0,stream>>>(bufU, g1, bufA);
    k_morph<<<tile,256,0,stream>>>(bufA, m1, mDo,mDy,mDx,mFy,mFx,mC1,mC2,mC3, bufC, 0);
    k_mix_wmma<<<(NB*OHW/16)/8,256,0,stream>>>(bufB, bufC, li, bufA);
    k_zero_stats<<<1,64,0,stream>>>(stats);
    k_bn_reduce<<<dim3(18,CC,NB),256,0,stream>>>(bufA, stats);
    k_bn_apply<<<nElemBlocks,256,0,stream>>>(bufA, bufU, stats, ga, be);
  }
  k_head<<<(NB*HWH+255)/256,256,0,stream>>>(bufU, final_w, out);
}